// VectorQuantizer_16329465659578
// MI455X (gfx1250) — compile-verified
//
#include <hip/hip_runtime.h>
#include <hip/hip_bf16.h>
#include <stdint.h>

// Problem constants
#define NE     8192      // num embeddings / channel dim
#define EDIM   256       // embedding dim
#define NPOS   4096      // B*H*W = 4*32*32
#define TAU_INV (1.0f/0.9f)
#define LOGT   (-9.010913347279288f)   // log(1/8192)
#define TINV   (1.0f/8192.0f)

// d_out layout (floats): [0,1048576) quantized, [1048576] kl, [1048577,...) one_hot
#define OUT_KL      1048576u
#define OUT_ONEHOT  1048577u
#define ONEHOT_N    33554432u

// workspace layout (4-byte units). wsf (float*) and wsi (int*) alias d_ws.
#define WS_ZM   0        // [4][4096] seg partial: running max of z
#define WS_ZS   16384    // [4][4096] seg partial: sum exp(z - max)
#define WS_XM   32768    // [4][4096] seg partial: running max of x
#define WS_XS   49152    // [4][4096] seg partial: sum exp(x - max)
#define WS_SX   65536    // [4][4096] seg partial: sum x
#define WS_ID   81920    // [4][4096] seg partial: argmax channel (int)
#define WS_IDX  98304    // [4096] final argmax (int)
#define WS_V    102400   // [4096] one-hot value (~1.0)
#define WS_KL   106496   // [128] per-(b,h) kl partial

// ---- CDNA5 async global->LDS copy (per-lane 16B, ASYNCcnt tracked) ----
__device__ __forceinline__ void async_b128(unsigned lds_off, const char* gaddr) {
  asm volatile("global_load_async_to_lds_b128 %0, %1, off"
               :: "v"(lds_off), "v"(gaddr) : "memory");
}
template <int N>
__device__ __forceinline__ void wait_async() {
  asm volatile("s_wait_asynccnt %0" :: "n"(N) : "memory");
}

// =====================================================================
// Pass 1: per (b,h,seg) block, 8 waves; wave j owns channels ch ≡ seg*8+j (mod 32).
// Streams x and gumbel tiles (8 channels x 32 w x 2 arrays = 2KB/buffer) via
// async-to-LDS double buffering, runs online softmax / argmax / lse / sum.
// =====================================================================
__global__ void __launch_bounds__(256) vq_pass1(const float* __restrict__ x,
                                                const float* __restrict__ g,
                                                float* wsf, int* wsi)
{
  __shared__ float stage[8][2][2][8][32];   // [wave][buf][arr(x/g)][ch][w] = 32 KB
  __shared__ float r_zm[8][32], r_zs[8][32], r_xm[8][32], r_xs[8][32], r_sx[8][32];
  __shared__ int   r_id[8][32];

  const int tid  = threadIdx.x;
  const int lane = tid & 31;          // lane == w
  const int wv   = tid >> 5;
  const int bh   = blockIdx.x >> 2;   // b*32 + h
  const int seg  = blockIdx.x & 3;
  const int b    = bh >> 5;
  const int h    = bh & 31;

  const int sub = lane & 7;           // 16B chunk within a 128B channel row
  const int cl4 = lane >> 3;          // which of 4 rows this lane services

  const unsigned sbase = (unsigned)(unsigned long long)&stage[0][0][0][0][0];
  const unsigned lwave = sbase + (unsigned)wv * 4096u + (unsigned)sub * 16u;

  const size_t rowoff = (size_t)b * (size_t)NE * 4096u + (size_t)h * 128u + (size_t)sub * 16u;
  const char* xb = (const char*)x + rowoff;
  const char* gb = (const char*)g + rowoff;
  const int chbase = seg * 8 + wv;    // channels chbase + 32*k, k = 0..255

  auto issue = [&](int t, int buf) {
    #pragma unroll
    for (int k = 0; k < 2; ++k) {
      int cl = k * 4 + cl4;                      // 0..7 channel within tile
      int ch = chbase + 32 * (t * 8 + cl);
      size_t cb = (size_t)ch * 4096u;
      unsigned ld = lwave + (unsigned)buf * 2048u + (unsigned)cl * 128u;
      async_b128(ld,         xb + cb);
      async_b128(ld + 1024u, gb + cb);
    }
  };

  float zm = -1e30f, zs = 0.f, xm = -1e30f, xs = 0.f, sx = 0.f;
  int id = 0;

  issue(0, 0);
  for (int t = 0; t < 32; ++t) {
    const int buf = t & 1;
    if (t + 1 < 32) { issue(t + 1, buf ^ 1); wait_async<4>(); }
    else            { wait_async<0>(); }
    #pragma unroll
    for (int c = 0; c < 8; ++c) {
      float xv = stage[wv][buf][0][c][lane];
      float gv = stage[wv][buf][1][c][lane];
      int   ch = chbase + 32 * (t * 8 + c);
      float z  = (xv + gv) * TAU_INV;
      float nm = fmaxf(zm, z);
      zs = zs * __expf(zm - nm) + __expf(z - nm);
      if (z > zm) id = ch;
      zm = nm;
      float nx = fmaxf(xm, xv);
      xs = xs * __expf(xm - nx) + __expf(xv - nx);
      xm = nx;
      sx += xv;
    }
  }

  r_zm[wv][lane] = zm; r_zs[wv][lane] = zs; r_id[wv][lane] = id;
  r_xm[wv][lane] = xm; r_xs[wv][lane] = xs; r_sx[wv][lane] = sx;
  __syncthreads();

  if (tid < 32) {   // wave 0: merge 8 wave-partials for position (bh, lane)
    float czm = -1e30f, czs = 0.f, cxm = -1e30f, cxs = 0.f, csx = 0.f;
    int cid = 0;
    #pragma unroll
    for (int j = 0; j < 8; ++j) {
      float m = r_zm[j][lane], s = r_zs[j][lane];
      float nm = fmaxf(czm, m);
      czs = czs * __expf(czm - nm) + s * __expf(m - nm);
      if (m > czm) cid = r_id[j][lane];
      czm = nm;
      float mx = r_xm[j][lane], se = r_xs[j][lane];
      float nx = fmaxf(cxm, mx);
      cxs = cxs * __expf(cxm - nx) + se * __expf(mx - nx);
      cxm = nx;
      csx += r_sx[j][lane];
    }
    int sidx = seg * NPOS + bh * 32 + lane;
    wsf[WS_ZM + sidx] = czm; wsf[WS_ZS + sidx] = czs;
    wsf[WS_XM + sidx] = cxm; wsf[WS_XS + sidx] = cxs;
    wsf[WS_SX + sidx] = csx; wsi[WS_ID + sidx] = cid;
  }
}

// =====================================================================
// Pass 1b: merge the 4 channel segments per position; emit idx, one-hot value v,
// and per-(b,h) kl partial (wave shfl reduction, deterministic).
// =====================================================================
__global__ void __launch_bounds__(32) vq_pass1b(float* wsf, int* wsi)
{
  const int lane = threadIdx.x;
  const int pos  = blockIdx.x * 32 + lane;

  float zm = -1e30f, zs = 0.f, xm = -1e30f, xs = 0.f, sx = 0.f;
  int id = 0;
  #pragma unroll
  for (int s = 0; s < 4; ++s) {
    int si = s * NPOS + pos;
    float m = wsf[WS_ZM + si], ss = wsf[WS_ZS + si];
    float nm = fmaxf(zm, m);
    zs = zs * __expf(zm - nm) + ss * __expf(m - nm);
    if (m > zm) id = wsi[WS_ID + si];
    zm = nm;
    float mx = wsf[WS_XM + si], se = wsf[WS_XS + si];
    float nx = fmaxf(xm, mx);
    xs = xs * __expf(xm - nx) + se * __expf(mx - nx);
    xm = nx;
    sx += wsf[WS_SX + si];
  }

  float ys = 1.0f / zs;              // softmax value at the argmax channel
  float v  = (1.0f - ys) + ys;       // straight-through forward value (== reference fp order)
  wsi[WS_IDX + pos] = id;
  wsf[WS_V   + pos] = v;

  float lse = xm + __logf(xs);       // logsumexp of x over channels
  float kl  = LOGT + lse - sx * TINV; // Σ_n t*(log t - log_qy) for this position
  for (int o = 16; o; o >>= 1) kl += __shfl_xor(kl, o, 32);
  if (lane == 0) wsf[WS_KL + blockIdx.x] = kl;
}

// =====================================================================
// KL reduce: sum the 128 per-(b,h) partials with two chained f32 WMMAs.
// A = all-ones (16x4)  =>  D[0][j] = column sums of B; shfl-add cols 0..15.
// =====================================================================
typedef float v2f __attribute__((ext_vector_type(2)));
typedef float v8f __attribute__((ext_vector_type(8)));

__global__ void __launch_bounds__(32) vq_klred(const float* wsf, float* out)
{
  const int lane = threadIdx.x;
  v2f a; a.x = 1.0f; a.y = 1.0f;                 // ones 16x4 A-matrix
  v2f b0, b1;                                    // any bijective packing works: D sums all of B
  b0.x = wsf[WS_KL + lane];      b0.y = wsf[WS_KL + 32 + lane];
  b1.x = wsf[WS_KL + 64 + lane]; b1.y = wsf[WS_KL + 96 + lane];
  v8f c = {};
  c = __builtin_amdgcn_wmma_f32_16x16x4_f32(false, a, false, b0, (short)0, c, false, false);
  c = __builtin_amdgcn_wmma_f32_16x16x4_f32(false, a, false, b1, (short)0, c, false, false);
  float r = c[0];                                 // row 0 (lanes 0-15) = per-column sums
  for (int o = 8; o; o >>= 1) r += __shfl_xor(r, o, 32);
  if (lane == 0) out[OUT_KL] = 0.0625f * r;       // COMMITMENT_COST(0.25) / B(4)
}

// =====================================================================
// Fill: quantized[b,d,h,w] = v * emb[idx, d] (coalesced stores over w; emb rows
// are L2-resident), plus scatter of the 4096 one-hot values (after memset zero).
// =====================================================================
__global__ void __launch_bounds__(256) vq_fill(const float* __restrict__ emb,
                                               const int* wsi, const float* wsf,
                                               float* out)
{
  const int lane = threadIdx.x & 31;   // w
  const int wv   = threadIdx.x >> 5;
  const int bh   = blockIdx.x;
  const int b    = bh >> 5;
  const int h    = bh & 31;
  const int pos  = bh * 32 + lane;

  const int   idx = wsi[WS_IDX + pos];
  const float v   = wsf[WS_V   + pos];

  if (wv == 0) {
    out[OUT_ONEHOT + (((size_t)b * NE + (size_t)idx) * 1024u + (size_t)h * 32u + (size_t)lane)] = v;
  }
  const float* row = emb + (size_t)idx * EDIM;
  #pragma unroll 4
  for (int d = wv; d < EDIM; d += 8) {
    out[(((size_t)b * EDIM + (size_t)d) * 32u + (size_t)h) * 32u + (size_t)lane] = v * row[d];
  }
}

// =====================================================================
extern "C" void kernel_launch(void* const* d_in, const int* in_sizes, int n_in,
                              void* d_out, int out_size, void* d_ws, size_t ws_size,
                              hipStream_t stream)
{
  const float* x   = (const float*)d_in[0];   // [4,8192,32,32]
  const float* emb = (const float*)d_in[1];   // [8192,256]
  const float* g   = (const float*)d_in[2];   // [4,8192,32,32]
  float* out = (float*)d_out;
  float* wsf = (float*)d_ws;
  int*   wsi = (int*)d_ws;

  // zero the one_hot region (134 MB) — capture-legal async memset at full bandwidth
  hipMemsetAsync(out + OUT_ONEHOT, 0, (size_t)ONEHOT_N * 4u, stream);

  vq_pass1 <<<512, 256, 0, stream>>>(x, g, wsf, wsi);
  vq_pass1b<<<128,  32, 0, stream>>>(wsf, wsi);
  vq_klred <<<  1,  32, 0, stream>>>(wsf, out);
  vq_fill  <<<128, 256, 0, stream>>>(emb, wsi, wsf, out);
}